// CRF_54674933678498
// MI455X (gfx1250) — compile-verified
//
#include <hip/hip_runtime.h>
#include <stdint.h>

#define BB 1024
#define SS 512
#define TT 64
#define TPAD 68          // padded trans rows in LDS (startup only)
#define WAVES 8
#define NEGV -10000.0f

// ---------------------------------------------------------------------------
// Kernel 1: Viterbi forward. One wave32 per batch element; lane l owns next-
// tags l and l+32, with both transition rows held in VGPRs (occupancy is
// batch-bound at ~1 wave/SIMD, so registers are free). Feats are streamed
// HBM -> LDS with the gfx1250 async-to-LDS engine (ASYNCcnt), depth-2
// pipelined. Argmax uses 4 independent 16-wide chains per tag to break the
// serial cndmask dependency chain; ordered-chunk strict-> merge preserves
// jnp.argmax first-occurrence semantics.
// ---------------------------------------------------------------------------
__global__ __launch_bounds__(256) void viterbi_fwd_kernel(
    const float* __restrict__ feats,      // [B][S][T]
    const float* __restrict__ trans,      // [T][T]  (row = next, col = prev)
    float* __restrict__ path_score,       // [B] -> d_out[0..B)
    uint8_t* __restrict__ bptr,           // [B][S][T] packed backpointers (ws)
    int* __restrict__ best_tag)           // [B] (ws tail)
{
    __shared__ float sTrans[TT * TPAD];       // 17408 B
    __shared__ float sFwd[WAVES * TT];        //  2048 B
    __shared__ float sFeat[WAVES * 2 * TT];   //  4096 B, double buffer per wave

    const int tid = threadIdx.x;
    const int w   = tid >> 5;
    const int l   = tid & 31;
    const int b   = blockIdx.x * WAVES + w;

    for (int i = tid; i < TT * TT; i += 256)
        sTrans[(i >> 6) * TPAD + (i & 63)] = trans[i];

    sFwd[w * TT + l]      = (l      == TT - 2) ? 0.0f : NEGV;   // START = 62
    sFwd[w * TT + l + 32] = (l + 32 == TT - 2) ? 0.0f : NEGV;
    __syncthreads();

    // Pull this lane's two transition rows into VGPRs (one-time DS cost).
    float ta[TT], tb[TT];
    #pragma unroll
    for (int p = 0; p < TT; ++p) {
        ta[p] = sTrans[l * TPAD + p];
        tb[p] = sTrans[(l + 32) * TPAD + p];
    }

    // Async feat streaming (feats total 128 MB < 4 GB -> u32 voffset OK).
    const uint32_t featBase = (uint32_t)((size_t)b * SS * TT * 4);
    const uint32_t ldsBase  = (uint32_t)(uintptr_t)&sFeat[w * 2 * TT];
    const unsigned long long gbase = (unsigned long long)(uintptr_t)feats;

    auto issue_row = [&](int r) {   // lane l loads dwords 2l,2l+1 of row r
        uint32_t voff = featBase + (uint32_t)r * (TT * 4) + (uint32_t)l * 8u;
        uint32_t lofs = ldsBase + (uint32_t)(r & 1) * (TT * 4) + (uint32_t)l * 8u;
        asm volatile("global_load_async_to_lds_b64 %0, %1, %2"
                     :: "v"(lofs), "v"(voff), "s"(gbase) : "memory");
    };

    issue_row(0);
    issue_row(1);

    float* fw = &sFwd[w * TT];
    // packed backpointers: byte 2l <- tag l, byte 2l+1 <- tag l+32
    uint16_t* bp16 = (uint16_t*)(bptr + (size_t)b * SS * TT) + l;

    for (int s = 0; s < SS; ++s) {
        // oldest outstanding async load (row s) done once cnt<=1 (in-order)
        asm volatile("s_wait_asynccnt 0x1" ::: "memory");

        float m0[4], m1[4];
        int   i0[4], i1[4];
        #pragma unroll
        for (int c = 0; c < 4; ++c) {
            m0[c] = -3.0e38f; m1[c] = -3.0e38f; i0[c] = 0; i1[c] = 0;
        }
        #pragma unroll
        for (int c = 0; c < 4; ++c) {
            #pragma unroll
            for (int q = 0; q < 16; ++q) {
                const int p = c * 16 + q;
                float f  = fw[p];                 // LDS broadcast read
                float v0 = f + ta[p];
                float v1 = f + tb[p];
                if (v0 > m0[c]) { m0[c] = v0; i0[c] = p; }  // strict '>': first max
                if (v1 > m1[c]) { m1[c] = v1; i1[c] = p; }
            }
        }
        // ordered-chunk merge, strict '>' keeps lowest index on ties
        float M0 = m0[0]; int I0 = i0[0];
        float M1 = m1[0]; int I1 = i1[0];
        #pragma unroll
        for (int c = 1; c < 4; ++c) {
            if (m0[c] > M0) { M0 = m0[c]; I0 = i0[c]; }
            if (m1[c] > M1) { M1 = m1[c]; I1 = i1[c]; }
        }

        const float* fb = &sFeat[(w * 2 + (s & 1)) * TT];
        float ft0 = fb[l];
        float ft1 = fb[l + 32];

        bp16[s * 32] = (uint16_t)((uint32_t)I0 | ((uint32_t)I1 << 8));

        // wave lockstep + in-order DS: all broadcast reads above precede these
        fw[l]      = M0 + ft0;
        fw[l + 32] = M1 + ft1;

        int r = s + 2;
        if (r > SS - 1) r = SS - 1;
        issue_row(r);                 // refill the buffer just consumed
    }

    // terminal: fwd + transitions[END=63][tag], wave-wide argmax (low tag wins ties)
    float v0 = fw[l]      + sTrans[(TT - 1) * TPAD + l];
    float v1 = fw[l + 32] + sTrans[(TT - 1) * TPAD + l + 32];
    float bv;
    int   bt;
    if (v1 > v0) { bv = v1; bt = l + 32; } else { bv = v0; bt = l; }
    #pragma unroll
    for (int off = 16; off > 0; off >>= 1) {
        float ov = __shfl_xor(bv, off, 32);
        int   ot = __shfl_xor(bt, off, 32);
        if (ov > bv || (ov == bv && ot < bt)) { bv = ov; bt = ot; }
    }
    if (l == 0) {
        path_score[b] = bv;
        best_tag[b]   = bt;
    }
}

// ---------------------------------------------------------------------------
// Kernel 2: backtrace. One thread per batch; the 32 MB backpointer array is
// L2-resident (192 MB L2). Packed layout remap: tag t lives at byte
// ((t&31)<<1) | (t>>5) within each 64-byte row.
// ---------------------------------------------------------------------------
__global__ __launch_bounds__(256) void viterbi_bt_kernel(
    const uint8_t* __restrict__ bptr,     // [B][S][T] packed
    const int* __restrict__ best_tag,     // [B]
    float* __restrict__ out_path)         // [B][S] -> d_out[B ..)
{
    int b = blockIdx.x * blockDim.x + threadIdx.x;
    if (b >= BB) return;
    int tag = best_tag[b];
    const uint8_t* bp = bptr + (size_t)b * SS * TT;
    float* row = out_path + (size_t)b * SS;
    for (int s = SS - 1; s >= 0; --s) {
        row[s] = (float)tag;                                  // ys[s] = carry
        int pos = ((tag & 31) << 1) | (tag >> 5);
        tag = bp[(size_t)s * TT + pos];
    }
}

extern "C" void kernel_launch(void* const* d_in, const int* in_sizes, int n_in,
                              void* d_out, int out_size, void* d_ws, size_t ws_size,
                              hipStream_t stream) {
    const float* feats = (const float*)d_in[0];   // [1024,512,64] f32
    const float* trans = (const float*)d_in[1];   // [64,64] f32
    float* out = (float*)d_out;                   // [0..B): path_score, [B..): best_path

    uint8_t* bptr = (uint8_t*)d_ws;                                   // 32 MB
    int* btag = (int*)((uint8_t*)d_ws + (size_t)BB * SS * TT);        // 4 KB

    viterbi_fwd_kernel<<<BB / WAVES, 256, 0, stream>>>(feats, trans, out, bptr, btag);
    viterbi_bt_kernel<<<(BB + 255) / 256, 256, 0, stream>>>(bptr, btag, out + BB);
}